// MultiHeadAttention_53437983097148
// MI455X (gfx1250) — compile-verified
//
#include <hip/hip_runtime.h>
#include <hip/hip_bf16.h>

typedef __attribute__((ext_vector_type(2))) float v2f;
typedef __attribute__((ext_vector_type(8))) float v8f;

#define C_DIM 512
#define S_DIM 1024
#define HD    64
#define QSZ   (8 * S_DIM * C_DIM)   // floats per Q/K/V tensor (4,194,304)

// ---- CDNA5 async global->LDS helpers (ASYNCcnt-tracked DMA, bypasses VGPRs)
__device__ __forceinline__ void async_tile4k(unsigned lds_addr,
                                             const float* gbase,
                                             unsigned voff) {
  // copies 4KB: 128 threads x 16B x 2; inst offset applies to BOTH lds & mem addr
  asm volatile("global_load_async_to_lds_b128 %0, %1, %2 offset:0"
               :: "v"(lds_addr), "v"(voff), "s"(gbase) : "memory");
  asm volatile("global_load_async_to_lds_b128 %0, %1, %2 offset:2048"
               :: "v"(lds_addr), "v"(voff), "s"(gbase) : "memory");
}
__device__ __forceinline__ void wait_async0() {
  asm volatile("s_wait_asynccnt 0x0" ::: "memory");
}

// ---------------------------------------------------------------------------
// Stage 1: QKV projection.  out[ty][b][s][co] = sum_c x[b][c][s]*W[co][c] + b
// Block = 4 waves.  Each wave: 16(M=s) x 64(N=co) tile (4 accumulators), so
// one A-fragment feeds 4 WMMAs -> ~2.5 loaded dwords per v_wmma.
// ---------------------------------------------------------------------------
__global__ __launch_bounds__(128) void qkv_proj_kernel(
    const float* __restrict__ x,
    const float* __restrict__ Wq, const float* __restrict__ bq,
    const float* __restrict__ Wk, const float* __restrict__ bk,
    const float* __restrict__ Wv, const float* __restrict__ bv,
    float* __restrict__ qkv)
{
  const int lane = threadIdx.x & 31;
  const int wave = threadIdx.x >> 5;
  const int half = lane >> 4;     // lanes 16-31 hold K+2,K+3 in A/B frags
  const int l16  = lane & 15;

  const int bz = blockIdx.z;      // 24 = 8 batches * 3 weight types
  const int b  = bz / 3;
  const int ty = bz % 3;
  const float* W    = (ty == 0) ? Wq : (ty == 1) ? Wk : Wv;
  const float* bias = (ty == 0) ? bq : (ty == 1) ? bk : bv;

  const float* xb   = x   + (size_t)b * C_DIM * S_DIM;        // [C][S]
  float*       outp = qkv + (size_t)ty * QSZ + (size_t)b * S_DIM * C_DIM;

  const int m0 = blockIdx.x * 64 + wave * 16; // token tile (per wave)
  const int n0 = blockIdx.y * 64;             // out-channel tile (4 chunks)

  v8f acc[4] = {};
  const int m = m0 + l16;
  const float* w0 = W + (size_t)(n0 +  0 + l16) * C_DIM;
  const float* w1 = W + (size_t)(n0 + 16 + l16) * C_DIM;
  const float* w2 = W + (size_t)(n0 + 32 + l16) * C_DIM;
  const float* w3 = W + (size_t)(n0 + 48 + l16) * C_DIM;

  #pragma unroll 4
  for (int k0 = 0; k0 < C_DIM; k0 += 4) {
    const int ka = k0 + 2 * half;
    // A[m][k] = t[s][c] = xb[c*S + s]  (coalesced across lanes)
    v2f a;
    a.x = xb[(size_t)ka * S_DIM + m];
    a.y = xb[(size_t)(ka + 1) * S_DIM + m];
    // B[k][n] = W[n][k] : two consecutive floats along each W row -> b64
    v2f bf0 = *(const v2f*)(w0 + ka);
    v2f bf1 = *(const v2f*)(w1 + ka);
    v2f bf2 = *(const v2f*)(w2 + ka);
    v2f bf3 = *(const v2f*)(w3 + ka);
    acc[0] = __builtin_amdgcn_wmma_f32_16x16x4_f32(false, a, false, bf0, (short)0, acc[0], false, false);
    acc[1] = __builtin_amdgcn_wmma_f32_16x16x4_f32(false, a, false, bf1, (short)0, acc[1], false, false);
    acc[2] = __builtin_amdgcn_wmma_f32_16x16x4_f32(false, a, false, bf2, (short)0, acc[2], false, false);
    acc[3] = __builtin_amdgcn_wmma_f32_16x16x4_f32(false, a, false, bf3, (short)0, acc[3], false, false);
  }

  #pragma unroll
  for (int nc = 0; nc < 4; ++nc) {
    const float bb = bias[n0 + nc * 16 + l16];
    #pragma unroll
    for (int r = 0; r < 8; ++r) {
      const int row = m0 + r + 8 * half;      // C-layout: M = r + 8*(lane/16)
      outp[(size_t)row * C_DIM + n0 + nc * 16 + l16] = acc[nc][r] + bb;
    }
  }
}

// ---------------------------------------------------------------------------
// Stage 2: flash-attention per "bugged-reshape" group g (64 groups, each a
// contiguous [1024 x 64] slab of the [B,S,C] QKV layout).
// Block = 4 waves, same group, 4 consecutive 16-query tiles.  K/V tiles are
// double-buffered in LDS via GLOBAL_LOAD_ASYNC_TO_LDS_B128 so the DMA of
// tile kt+1 overlaps the 32 WMMAs + softmax of tile kt.
// ---------------------------------------------------------------------------
__global__ __launch_bounds__(128) void attn_kernel(
    const float* __restrict__ qkv, float* __restrict__ out)
{
  __shared__ float ldsK[2][16 * HD];   // 2 x 4KB
  __shared__ float ldsV[2][16 * HD];   // 2 x 4KB
  __shared__ float ldsP[4][16 * 16];   // per-wave transpose scratch

  const int tid  = threadIdx.x;
  const int lane = tid & 31;
  const int wave = tid >> 5;
  const int half = lane >> 4;
  const int l16  = lane & 15;

  const int g  = blockIdx.y;
  const int qt = blockIdx.x * 4 + wave;
  const int m0 = qt * 16;

  const float* Qg = qkv +                  (size_t)g * (S_DIM * HD);
  const float* Kg = qkv + (size_t)QSZ     + (size_t)g * (S_DIM * HD);
  const float* Vg = qkv + (size_t)QSZ * 2 + (size_t)g * (S_DIM * HD);

  const unsigned lk0 = (unsigned)(size_t)&ldsK[0][0] + tid * 16u;
  const unsigned lk1 = (unsigned)(size_t)&ldsK[1][0] + tid * 16u;
  const unsigned lv0 = (unsigned)(size_t)&ldsV[0][0] + tid * 16u;
  const unsigned lv1 = (unsigned)(size_t)&ldsV[1][0] + tid * 16u;

  // Q tile as 16 A-fragments (K-steps of 4 over hd=64), pre-scaled by 1/sqrt(64)
  v2f aQ[16];
  {
    const float* qrow = Qg + (size_t)(m0 + l16) * HD + 2 * half;
    #pragma unroll
    for (int j = 0; j < 16; ++j) {
      v2f t = *(const v2f*)(qrow + 4 * j);
      aQ[j].x = t.x * 0.125f;
      aQ[j].y = t.y * 0.125f;
    }
  }

  v8f acc[4] = {};           // 16 x 64 output accumulator (4 N-chunks of 16)
  float mrow[8], lrow[8];    // per-row online-softmax stats (row = r + 8*half)
  #pragma unroll
  for (int r = 0; r < 8; ++r) { mrow[r] = -1e30f; lrow[r] = 0.0f; }

  // prime the pipeline: DMA tile 0 into buffer 0
  async_tile4k(lk0, Kg, tid * 16u);
  async_tile4k(lv0, Vg, tid * 16u);

  for (int kt = 0; kt < 64; ++kt) {
    const int buf = kt & 1;
    wait_async0();     // this wave's DMA for tile kt has landed
    __syncthreads();   // everyone's tile-kt DMA landed; tile kt-1 fully consumed

    if (kt + 1 < 64) { // overlap: DMA tile kt+1 into the other buffer
      const unsigned voff = (unsigned)(kt + 1) * (16u * HD * 4u) + tid * 16u;
      async_tile4k(buf ? lk0 : lk1, Kg, voff);
      async_tile4k(buf ? lv0 : lv1, Vg, voff);
    }

    const float* Kt = &ldsK[buf][0];
    const float* Vt = &ldsV[buf][0];

    // S = (Q/sqrt(hd)) * K^T : 16 WMMAs, B[k][n] = K[key=n][d=k]
    v8f s = {};
    #pragma unroll
    for (int j = 0; j < 16; ++j) {
      v2f bk_ = *(const v2f*)(Kt + l16 * HD + 4 * j + 2 * half);
      s = __builtin_amdgcn_wmma_f32_16x16x4_f32(
          false, aQ[j], false, bk_, (short)0, s, false, false);
    }

    // Online softmax: row reductions across the 16 lanes of each half
    #pragma unroll
    for (int r = 0; r < 8; ++r) {
      float v = s[r];
      v = fmaxf(v, __shfl_xor(v, 1, 32));
      v = fmaxf(v, __shfl_xor(v, 2, 32));
      v = fmaxf(v, __shfl_xor(v, 4, 32));
      v = fmaxf(v, __shfl_xor(v, 8, 32));
      const float mnew = fmaxf(mrow[r], v);
      const float corr = __expf(mrow[r] - mnew);
      mrow[r] = mnew;
      lrow[r] *= corr;
      #pragma unroll
      for (int nc = 0; nc < 4; ++nc) acc[nc][r] *= corr;
      const float p = __expf(s[r] - mnew);
      s[r] = p;
      float ps = p;
      ps += __shfl_xor(ps, 1, 32);
      ps += __shfl_xor(ps, 2, 32);
      ps += __shfl_xor(ps, 4, 32);
      ps += __shfl_xor(ps, 8, 32);
      lrow[r] += ps;
    }

    // C-layout -> A-layout transpose of P through per-wave LDS scratch
    float* pl = &ldsP[wave][0];
    #pragma unroll
    for (int r = 0; r < 8; ++r)
      pl[(r + 8 * half) * 16 + l16] = s[r];
    v2f aP[4];
    #pragma unroll
    for (int j2 = 0; j2 < 4; ++j2)
      aP[j2] = *(const v2f*)(pl + l16 * 16 + 4 * j2 + 2 * half);

    // acc += P * V : B[k][n] = V[key=k][d = nc*16+n]
    #pragma unroll
    for (int nc = 0; nc < 4; ++nc) {
      #pragma unroll
      for (int j2 = 0; j2 < 4; ++j2) {
        const int k = 4 * j2 + 2 * half;
        v2f bv_;
        bv_.x = Vt[k * HD + nc * 16 + l16];
        bv_.y = Vt[(k + 1) * HD + nc * 16 + l16];
        acc[nc] = __builtin_amdgcn_wmma_f32_16x16x4_f32(
            false, aP[j2], false, bv_, (short)0, acc[nc], false, false);
      }
    }
  }

  // Epilogue: normalize and write with inverse of the bugged reshape:
  // token i -> (s_spatial = (g%8)*128 + i/8 , c = (i%8)*64 + d), out is [B,C,S]
  const int b  = g >> 3;
  const int gs = g & 7;
  float inv[8];
  #pragma unroll
  for (int r = 0; r < 8; ++r) inv[r] = 1.0f / lrow[r];
  #pragma unroll
  for (int nc = 0; nc < 4; ++nc) {
    #pragma unroll
    for (int r = 0; r < 8; ++r) {
      const int i    = m0 + r + 8 * half;
      const int d    = nc * 16 + l16;
      const int c    = (i & 7) * 64 + d;
      const int srow = gs * 128 + (i >> 3);
      out[(size_t)b * (C_DIM * S_DIM) + (size_t)c * S_DIM + srow] =
          acc[nc][r] * inv[r];
    }
  }
}

// ---------------------------------------------------------------------------
extern "C" void kernel_launch(void* const* d_in, const int* in_sizes, int n_in,
                              void* d_out, int out_size, void* d_ws, size_t ws_size,
                              hipStream_t stream) {
  const float* x  = (const float*)d_in[0];
  const float* Wq = (const float*)d_in[1];
  const float* bq = (const float*)d_in[2];
  const float* Wk = (const float*)d_in[3];
  const float* bk = (const float*)d_in[4];
  const float* Wv = (const float*)d_in[5];
  const float* bv = (const float*)d_in[6];
  float* out = (float*)d_out;
  float* qkv = (float*)d_ws;   // needs 3 * 4,194,304 floats = 50.3 MB

  dim3 g1(16, 8, 24);          // (m-tiles of 64, n-tiles of 64, batch*3)
  qkv_proj_kernel<<<g1, 128, 0, stream>>>(x, Wq, bq, Wk, bk, Wv, bv, qkv);

  dim3 g2(16, 64);             // (q-tile quads, groups)
  attn_kernel<<<g2, 128, 0, stream>>>(qkv, out);
}